// LlamaAttention_heavy_hitter_65970697667352
// MI455X (gfx1250) — compile-verified
//
#include <hip/hip_runtime.h>
#include <math.h>

#define B 4
#define NH 32
#define HD 128
#define HID 4096
#define S_PAST 8192
#define SKV 8193          // past + 1 new token
#define BH (B * NH)
#define HEAVY 2048
#define OUTLIER 16        // HD / GROUP_FACTOR

typedef __attribute__((ext_vector_type(2))) float v2f;
typedef __attribute__((ext_vector_type(8))) float v8f;

// ---------------------------------------------------------------------------
// WMMA fp32 GEMM: out[b][o] = sum_i A[b][i] * W[o][i], b<4 (padded to M=16).
// One wave per 16-wide output tile; V_WMMA_F32_16X16X4_F32, K step = 4.
// A 16x4 layout: lanes 0-15 -> M=0..15 {K=0,K=1}; lanes 16-31 -> {K=2,K=3}.
// B 4x16 layout: lanes 0-15 -> N, rows {K=0,K=1}; lanes 16-31 -> {K=2,K=3}.
// C 16x16 layout: VGPR j holds row j (lanes 0-15) / row j+8 (lanes 16-31).
// ---------------------------------------------------------------------------
__global__ __launch_bounds__(256) void gemm16_wmma(const float* __restrict__ A,
                                                   const float* __restrict__ W,
                                                   float* __restrict__ out) {
  const int wave = threadIdx.x >> 5;
  const int L    = threadIdx.x & 31;
  const int o0   = (blockIdx.x * 8 + wave) * 16;
  const int half = L >> 4;
  const int mn   = L & 15;                       // M for A side, N for B side
  const float amask = (mn < 4) ? 1.0f : 0.0f;    // pad rows 4..15 with zeros
  const float* arow = A + (size_t)(mn < 4 ? mn : 0) * HID;
  const float* wrow = W + (size_t)(o0 + mn) * HID;
  const int koff = half * 2;

  v8f c = {0.f, 0.f, 0.f, 0.f, 0.f, 0.f, 0.f, 0.f};
  for (int k0 = 0; k0 < HID; k0 += 4) {
    v2f a, bv;
    a.x  = arow[k0 + koff]     * amask;
    a.y  = arow[k0 + koff + 1] * amask;
    bv.x = wrow[k0 + koff];
    bv.y = wrow[k0 + koff + 1];
    c = __builtin_amdgcn_wmma_f32_16x16x4_f32(false, a, false, bv,
                                              (short)0, c, false, false);
  }
  if (half == 0) {
#pragma unroll
    for (int j = 0; j < 4; ++j)                  // only rows 0..3 are real
      out[(size_t)j * HID + o0 + mn] = c[j];
  }
}

// ---------------------------------------------------------------------------
// RoPE: x*cos + rotate_half(x)*sin  (cos/sin depend only on (b, d))
// ---------------------------------------------------------------------------
__global__ __launch_bounds__(256) void rope_kernel(const float* __restrict__ qin,
                                                   const float* __restrict__ kin,
                                                   const float* __restrict__ cosv,
                                                   const float* __restrict__ sinv,
                                                   float* __restrict__ qout,
                                                   float* __restrict__ kout) {
  int idx = blockIdx.x * blockDim.x + threadIdx.x;
  if (idx >= 2 * B * HID) return;
  int which = idx / (B * HID);
  int e = idx - which * (B * HID);
  int b = e >> 12;               // / HID
  int c = e & (HID - 1);
  int d = c & (HD - 1);
  const float* src = which ? kin : qin;
  float x = src[e];
  float other = (d < HD / 2) ? -src[e + HD / 2] : src[e - HD / 2];
  float r = x * cosv[b * HD + d] + other * sinv[b * HD + d];
  (which ? kout : qout)[e] = r;
}

// ---------------------------------------------------------------------------
// pseudo_quantize over 16 values (LABEL_BITS=4). rintf == round-half-even
// matches jnp.round.
// ---------------------------------------------------------------------------
__device__ inline void quantize16(const float* in, float* out) {
  float mn = in[0], mx = in[0];
#pragma unroll
  for (int j = 1; j < OUTLIER; ++j) { mn = fminf(mn, in[j]); mx = fmaxf(mx, in[j]); }
  float range = mx - mn;
  if (range == 0.0f) range = 1.0f;
  float scale = 15.0f / range;
#pragma unroll
  for (int j = 0; j < OUTLIER; ++j) {
    float qv = rintf((in[j] - mn) * scale);
    qv = fminf(fmaxf(qv, 0.0f), 15.0f);
    out[j] = qv / scale + mn;
  }
}

// ---------------------------------------------------------------------------
// Scores: attn[s] = <q, K_s>/sqrt(128); gattn[s] = <gq, quant(gk_s)>/sqrt(16)
// One block per (b,h); streams past_key once.
// ---------------------------------------------------------------------------
__global__ __launch_bounds__(256) void score_kernel(const float* __restrict__ qr,
                                                    const float* __restrict__ kr,
                                                    const float* __restrict__ past_key,
                                                    const int* __restrict__ sorted_channel,
                                                    float* __restrict__ attn,
                                                    float* __restrict__ gattn) {
  const int bh = blockIdx.x;
  const int h  = bh & (NH - 1);
  const int t  = threadIdx.x;
  __shared__ float qv[HD];
  __shared__ float graw[OUTLIER];
  __shared__ float gq[OUTLIER];
  __shared__ int   sc[OUTLIER];
  if (t < HD) qv[t] = qr[(size_t)bh * HD + t];
  if (t < OUTLIER) sc[t] = sorted_channel[h * HD + t];
  __syncthreads();
  if (t < OUTLIER) graw[t] = qv[sc[t]];
  __syncthreads();
  if (t == 0) quantize16(graw, gq);
  __syncthreads();

  const float* krow_new = kr + (size_t)bh * HD;
  const float* kp       = past_key + (size_t)bh * S_PAST * HD;
  float* arow = attn  + (size_t)bh * SKV;
  float* grow = gattn + (size_t)bh * SKV;

  for (int s = t; s < SKV; s += 256) {
    const float* krow = (s < S_PAST) ? (kp + (size_t)s * HD) : krow_new;
    const float4* k4 = (const float4*)krow;
    const float4* q4 = (const float4*)qv;
    float acc = 0.0f;
#pragma unroll 4
    for (int i = 0; i < HD / 4; ++i) {
      float4 kk = k4[i]; float4 qq = q4[i];
      acc += kk.x * qq.x + kk.y * qq.y + kk.z * qq.z + kk.w * qq.w;
    }
    arow[s] = acc * 0.08838834764831845f;   // 1/sqrt(128)

    float gk[OUTLIER], gkq[OUTLIER];
#pragma unroll
    for (int j = 0; j < OUTLIER; ++j) gk[j] = krow[sc[j]];
    quantize16(gk, gkq);
    float gacc = 0.0f;
#pragma unroll
    for (int j = 0; j < OUTLIER; ++j) gacc += gkq[j] * gq[j];
    grow[s] = gacc * 0.25f;                 // 1/sqrt(16)
  }
}

// ---------------------------------------------------------------------------
// Exact k-th largest (k=2048) per row via 4-pass MSB radix select.
// ---------------------------------------------------------------------------
__device__ inline unsigned f2ord(float f) {
  unsigned u = __float_as_uint(f);
  return (u & 0x80000000u) ? ~u : (u | 0x80000000u);
}
__device__ inline float ord2f(unsigned u) {
  return (u & 0x80000000u) ? __uint_as_float(u & 0x7FFFFFFFu)
                           : __uint_as_float(~u);
}

__global__ __launch_bounds__(256) void kth_kernel(const float* __restrict__ gattn,
                                                  float* __restrict__ kth) {
  const int bh = blockIdx.x;
  const float* g = gattn + (size_t)bh * SKV;
  __shared__ unsigned hist[256];
  __shared__ unsigned s_prefix;
  __shared__ unsigned s_rem;
  if (threadIdx.x == 0) { s_prefix = 0u; s_rem = HEAVY; }
  __syncthreads();
  for (int shift = 24; shift >= 0; shift -= 8) {
    hist[threadIdx.x] = 0u;
    __syncthreads();
    unsigned pref = s_prefix;
    unsigned mask = (shift == 24) ? 0u : (0xFFFFFFFFu << (shift + 8));
    for (int s = threadIdx.x; s < SKV; s += 256) {
      unsigned key = f2ord(g[s]);
      if ((key & mask) == pref) atomicAdd(&hist[(key >> shift) & 0xFFu], 1u);
    }
    __syncthreads();
    if (threadIdx.x == 0) {
      unsigned r = s_rem, p = s_prefix;
      for (int bb = 255; bb >= 0; --bb) {
        unsigned cnt = hist[bb];
        if (cnt >= r) { s_prefix = p | ((unsigned)bb << shift); break; }
        r -= cnt;
      }
      s_rem = r;
    }
    __syncthreads();
  }
  if (threadIdx.x == 0) kth[bh] = ord2f(s_prefix);
}

// ---------------------------------------------------------------------------
// Masked softmax: keep gattn >= kth (ties kept, matching jnp.where >= kth).
// Overwrites attn with normalized weights.
// ---------------------------------------------------------------------------
__global__ __launch_bounds__(256) void softmax_kernel(float* __restrict__ attn,
                                                      const float* __restrict__ gattn,
                                                      const float* __restrict__ kth) {
  const int bh = blockIdx.x;
  const int t  = threadIdx.x;
  const float th = kth[bh];
  float* a = attn + (size_t)bh * SKV;
  const float* g = gattn + (size_t)bh * SKV;
  __shared__ float red[256];

  float m = -3.4e38f;
  for (int s = t; s < SKV; s += 256)
    if (g[s] >= th) m = fmaxf(m, a[s]);
  red[t] = m; __syncthreads();
  for (int off = 128; off > 0; off >>= 1) {
    if (t < off) red[t] = fmaxf(red[t], red[t + off]);
    __syncthreads();
  }
  const float M = red[0];
  __syncthreads();

  float sum = 0.0f;
  for (int s = t; s < SKV; s += 256)
    if (g[s] >= th) sum += expf(a[s] - M);
  red[t] = sum; __syncthreads();
  for (int off = 128; off > 0; off >>= 1) {
    if (t < off) red[t] += red[t + off];
    __syncthreads();
  }
  const float invS = 1.0f / red[0];
  __syncthreads();

  for (int s = t; s < SKV; s += 256)
    a[s] = (g[s] >= th) ? expf(a[s] - M) * invS : 0.0f;
}

// ---------------------------------------------------------------------------
// AV: ao[b, h*HD+d] = sum_s w[s] * V[s][d].  Streams past_value once.
// 256 threads: d = t&127, two s-parity groups reduced through LDS.
// ---------------------------------------------------------------------------
__global__ __launch_bounds__(256) void av_kernel(const float* __restrict__ w,
                                                 const float* __restrict__ past_value,
                                                 const float* __restrict__ vnew,
                                                 float* __restrict__ ao) {
  const int bh = blockIdx.x;
  const int t  = threadIdx.x;
  const int d  = t & (HD - 1);
  const int sg = t >> 7;
  const float* wr = w + (size_t)bh * SKV;
  const float* pv = past_value + (size_t)bh * S_PAST * HD;
  float acc = 0.0f;
  for (int s = sg; s < S_PAST; s += 2)
    acc += wr[s] * pv[(size_t)s * HD + d];
  if (sg == 0)
    acc += wr[S_PAST] * vnew[(size_t)bh * HD + d];
  __shared__ float red[256];
  red[t] = acc; __syncthreads();
  if (sg == 0)
    ao[(size_t)(bh >> 5) * HID + (bh & (NH - 1)) * HD + d] = red[d] + red[HD + d];
}

// ---------------------------------------------------------------------------
extern "C" void kernel_launch(void* const* d_in, const int* in_sizes, int n_in,
                              void* d_out, int out_size, void* d_ws, size_t ws_size,
                              hipStream_t stream) {
  const float* hs   = (const float*)d_in[0];   // hidden_states (B,1,HID)
  const float* pk   = (const float*)d_in[1];   // past_key  (B,NH,S,HD)
  const float* pv   = (const float*)d_in[2];   // past_value(B,NH,S,HD)
  const float* cosv = (const float*)d_in[3];   // (B,1,HD)
  const float* sinv = (const float*)d_in[4];   // (B,1,HD)
  const int*   sch  = (const int*)d_in[5];     // (NH,HD) int32
  const float* Wq   = (const float*)d_in[6];
  const float* Wk   = (const float*)d_in[7];
  const float* Wv   = (const float*)d_in[8];
  const float* Wo   = (const float*)d_in[9];
  float* out = (float*)d_out;

  float* ws   = (float*)d_ws;
  float* q    = ws;                         // B*HID
  float* k    = q   + (size_t)B * HID;
  float* v    = k   + (size_t)B * HID;
  float* qr   = v   + (size_t)B * HID;
  float* kr   = qr  + (size_t)B * HID;
  float* ao   = kr  + (size_t)B * HID;
  float* kth  = ao  + (size_t)B * HID;      // BH floats
  float* attn = kth + 128;                  // BH*SKV
  float* gatt = attn + (size_t)BH * SKV;    // BH*SKV

  const dim3 gemmGrid(HID / 16 / 8), gemmBlk(256);
  hipLaunchKernelGGL(gemm16_wmma, gemmGrid, gemmBlk, 0, stream, hs, Wq, q);
  hipLaunchKernelGGL(gemm16_wmma, gemmGrid, gemmBlk, 0, stream, hs, Wk, k);
  hipLaunchKernelGGL(gemm16_wmma, gemmGrid, gemmBlk, 0, stream, hs, Wv, v);

  hipLaunchKernelGGL(rope_kernel, dim3((2 * B * HID + 255) / 256), dim3(256), 0,
                     stream, q, k, cosv, sinv, qr, kr);

  hipLaunchKernelGGL(score_kernel, dim3(BH), dim3(256), 0, stream,
                     qr, kr, pk, sch, attn, gatt);

  hipLaunchKernelGGL(kth_kernel, dim3(BH), dim3(256), 0, stream, gatt, kth);

  hipLaunchKernelGGL(softmax_kernel, dim3(BH), dim3(256), 0, stream,
                     attn, gatt, kth);

  hipLaunchKernelGGL(av_kernel, dim3(BH), dim3(256), 0, stream,
                     attn, pv, v, ao);

  hipLaunchKernelGGL(gemm16_wmma, gemmGrid, gemmBlk, 0, stream, ao, Wo, out);
}